// Strict2_5DLoss_53979148976612
// MI455X (gfx1250) — compile-verified
//
#include <hip/hip_runtime.h>

typedef __attribute__((ext_vector_type(2))) float v2f;
typedef __attribute__((ext_vector_type(8))) float v8f;

#define NB    4
#define NGRP  64
#define HH    192
#define WW    192
#define NPIX  (HH * WW)
#define NC    4
#define KCAP  96
#define MAXC  4096
#define ETA_F 3.0f
#define POS_W 1.2f

__device__ __forceinline__ unsigned long long umin64(unsigned long long a, unsigned long long b) {
  return a < b ? a : b;
}

__device__ __forceinline__ float seg_dist(float px, float py, float Px, float Py, float Qx, float Qy) {
  float vx = Qx - Px, vy = Qy - Py;
  float wx = px - Px, wy = py - Py;
  float t = (wx * vx + wy * vy) / (vx * vx + vy * vy + 1e-9f);
  t = fminf(fmaxf(t, 0.0f), 1.0f);
  float dx = wx - t * vx, dy = wy - t * vy;
  return sqrtf(dx * dx + dy * dy + 1e-12f);
}

__device__ __forceinline__ float log_sigmoid(float x) {
  float ax = fabsf(x);
  return fminf(x, 0.0f) - log1pf(__expf(-ax));
}

// ---------------- K0: init cell keys + zero outputs ----------------
__global__ void __launch_bounds__(256) k_init(unsigned long long* cellkey, float* out) {
  int i = blockIdx.x * blockDim.x + threadIdx.x;
  if (i < NB * NPIX) cellkey[i] = ~0ULL;
  if (i < 3) out[i] = 0.0f;
}

// ---------------- K1: per-(b,g) distance + top-96 selection ----------------
// One 256-thread workgroup per (b,g). Edge-sign values d1,d2,d3 are affine in
// (px,py) and computed with V_WMMA_F32_16X16X4_F32 (A = edge coeff rows,
// B = pixel columns). Segment distances in f32 VALU. Positives compacted to
// LDS, then 96 iterative min-extractions on keys (dist_bits<<32 | n), which
// matches jax.lax.top_k ordering (asc dist, ties by low index).
__global__ void __launch_bounds__(256) k_select(const float* __restrict__ gt_pts,
                                                const int* __restrict__ stride_p,
                                                unsigned long long* __restrict__ sel_key,
                                                unsigned long long* __restrict__ cellkey) {
  __shared__ unsigned long long cand[MAXC];
  __shared__ unsigned long long red[256];
  __shared__ int cnt;

  int bg = blockIdx.x;
  int b = bg >> 6, g = bg & 63;
  float s = (float)stride_p[0];
  const float* tp = gt_pts + (size_t)(b * NGRP + g) * 6;
  float Ax = tp[0], Ay = tp[1], Bx = tp[2], By = tp[3], Cx = tp[4], Cy = tp[5];

  if (threadIdx.x == 0) cnt = 0;
  __syncthreads();

  // Edge e: d(px,py) = a*px + b*py + c   (a = Py-Qy, b = Qx-Px, c = Px*Qy - Py*Qx)
  // e0:(A,B)  e1:(B,C)  e2:(C,A)
  int lane = threadIdx.x & 31;
  int wave = threadIdx.x >> 5;
  int m = lane & 15;
  bool hiHalf = lane >= 16;

  float ca = 0.0f, cb = 0.0f, cc = 0.0f;
  if (m == 0) { ca = Ay - By; cb = Bx - Ax; cc = Ax * By - Ay * Bx; }
  else if (m == 1) { ca = By - Cy; cb = Cx - Bx; cc = Bx * Cy - By * Cx; }
  else if (m == 2) { ca = Cy - Ay; cb = Ax - Cx; cc = Cx * Ay - Cy * Ax; }

  v2f avec;  // A 16x4: lanes 0-15 {K0,K1}={a,b}; lanes 16-31 {K2,K3}={c,0}
  avec.x = hiHalf ? cc : ca;
  avec.y = hiHalf ? 0.0f : cb;
  v8f czero = {};

  for (int base = wave * 32; base < NPIX; base += 256) {
    // This wave handles pixels base..base+31; two WMMAs of 16 pixel columns.
    int na = base + m;
    int nb2 = base + 16 + m;
    float pax = ((float)(na % WW) + 0.5f) * s;
    float pay = ((float)(na / WW) + 0.5f) * s;
    float pbx = ((float)(nb2 % WW) + 0.5f) * s;
    float pby = ((float)(nb2 / WW) + 0.5f) * s;
    // B 4x16: v0 = row0 (lanes 0-15) | row2 (16-31); v1 = row1 | row3
    v2f bva, bvb;
    bva.x = hiHalf ? 1.0f : pax;  bva.y = hiHalf ? 0.0f : pay;
    bvb.x = hiHalf ? 1.0f : pbx;  bvb.y = hiHalf ? 0.0f : pby;

#if __has_builtin(__builtin_amdgcn_wmma_f32_16x16x4_f32)
    v8f da = __builtin_amdgcn_wmma_f32_16x16x4_f32(false, avec, false, bva, (short)0, czero, false, false);
    v8f db = __builtin_amdgcn_wmma_f32_16x16x4_f32(false, avec, false, bvb, (short)0, czero, false, false);
    // D rows 0..2 (v0..v2), lanes 0-15 = cols 0..15. Move tile-b results up.
    float d1b = __shfl_xor(db[0], 16, 32);
    float d2b = __shfl_xor(db[1], 16, 32);
    float d3b = __shfl_xor(db[2], 16, 32);
    float d1 = hiHalf ? d1b : da[0];
    float d2 = hiHalf ? d2b : da[1];
    float d3 = hiHalf ? d3b : da[2];
#else
    float pxl = hiHalf ? pbx : pax, pyl = hiHalf ? pby : pay; (void)czero;
    float d1 = (pxl - Bx) * (Ay - By) - (Ax - Bx) * (pyl - By);
    float d2 = (pxl - Cx) * (By - Cy) - (Bx - Cx) * (pyl - Cy);
    float d3 = (pxl - Ax) * (Cy - Ay) - (Cx - Ax) * (pyl - Ay);
#endif

    int n = base + lane;
    float px = ((float)(n % WW) + 0.5f) * s;
    float py = ((float)(n / WW) + 0.5f) * s;
    bool has_neg = (d1 < 0.0f) || (d2 < 0.0f) || (d3 < 0.0f);
    bool has_pos = (d1 > 0.0f) || (d2 > 0.0f) || (d3 > 0.0f);
    bool inside = !(has_neg && has_pos);
    float dist = fminf(seg_dist(px, py, Ax, Ay, Bx, By),
                 fminf(seg_dist(px, py, Bx, By, Cx, Cy),
                       seg_dist(px, py, Cx, Cy, Ax, Ay)));
    if (inside || (dist <= ETA_F)) {
      int slot = atomicAdd(&cnt, 1);
      if (slot < MAXC)
        cand[slot] = ((unsigned long long)__float_as_uint(dist) << 32) | (unsigned)n;
    }
  }
  __syncthreads();
  int M = cnt < MAXC ? cnt : MAXC;

  for (int k = 0; k < KCAP; ++k) {
    unsigned long long lm = ~0ULL;
    for (int i = threadIdx.x; i < M; i += 256) lm = umin64(lm, cand[i]);
    red[threadIdx.x] = lm;
    __syncthreads();
    for (int off = 128; off > 0; off >>= 1) {
      if (threadIdx.x < off) red[threadIdx.x] = umin64(red[threadIdx.x], red[threadIdx.x + off]);
      __syncthreads();
    }
    unsigned long long best = red[0];
    if (threadIdx.x == 0) sel_key[(size_t)bg * KCAP + k] = best;
    if (best != ~0ULL) {
      for (int i = threadIdx.x; i < M; i += 256)
        if (cand[i] == best) cand[i] = ~0ULL;   // keys unique (distinct n)
      if (threadIdx.x == 0) {
        unsigned n = (unsigned)best;
        unsigned long long ck = (best & 0xFFFFFFFF00000000ULL) | (unsigned)g;
        atomicMin(&cellkey[(size_t)b * NPIX + n], ck);
      }
    }
    __syncthreads();
  }
}

// ---------------- K2: objectness BCE + per-cell cross entropy ----------------
__global__ void __launch_bounds__(256) k_objcls(const float* __restrict__ pred_obj,
                                                const float* __restrict__ pred_cls,
                                                const int* __restrict__ gt_lbl,
                                                const unsigned long long* __restrict__ cellkey,
                                                float* __restrict__ out) {
  __shared__ float r0[256], r1[256];
  int gid = blockIdx.x * blockDim.x + threadIdx.x;
  float oL = 0.0f, cL = 0.0f;
  if (gid < NB * NPIX) {
    int b = gid / NPIX, n = gid % NPIX;
    unsigned long long ck = cellkey[gid];
    float x = pred_obj[gid];
    if (ck != ~0ULL) {
      oL = -(POS_W * log_sigmoid(x));
      int g = (int)(unsigned)ck;
      int tgt = gt_lbl[b * NGRP + g];
      float l0 = pred_cls[((size_t)(b * NC) + 0) * NPIX + n];
      float l1 = pred_cls[((size_t)(b * NC) + 1) * NPIX + n];
      float l2 = pred_cls[((size_t)(b * NC) + 2) * NPIX + n];
      float l3 = pred_cls[((size_t)(b * NC) + 3) * NPIX + n];
      float mx = fmaxf(fmaxf(l0, l1), fmaxf(l2, l3));
      float sm = __expf(l0 - mx) + __expf(l1 - mx) + __expf(l2 - mx) + __expf(l3 - mx);
      float lse = mx + __logf(sm);
      float picked = (tgt == 0) ? l0 : ((tgt == 1) ? l1 : ((tgt == 2) ? l2 : l3));
      cL = lse - picked;
    } else {
      oL = -log_sigmoid(-x);
    }
  }
  r0[threadIdx.x] = oL; r1[threadIdx.x] = cL;
  __syncthreads();
  for (int off = 128; off > 0; off >>= 1) {
    if (threadIdx.x < off) {
      r0[threadIdx.x] += r0[threadIdx.x + off];
      r1[threadIdx.x] += r1[threadIdx.x + off];
    }
    __syncthreads();
  }
  if (threadIdx.x == 0) { atomicAdd(&out[1], r0[0]); atomicAdd(&out[2], r1[0]); }
}

// ---------------- K3: regression (p0 L2 + 2x2 chamfer) over selected cells ----------------
__global__ void __launch_bounds__(256) k_reg(const float* __restrict__ pred_reg,
                                             const float* __restrict__ gt_pts,
                                             const int* __restrict__ stride_p,
                                             const unsigned long long* __restrict__ sel_key,
                                             float* __restrict__ out) {
  __shared__ float r0[256];
  int gid = blockIdx.x * blockDim.x + threadIdx.x;
  float acc = 0.0f;
  if (gid < NB * NGRP * KCAP) {
    unsigned long long key = sel_key[gid];
    if (key != ~0ULL) {
      float s = (float)stride_p[0];
      int bg = gid / KCAP;
      int b = bg >> 6, g = bg & 63;
      unsigned n = (unsigned)key;
      float ancx = ((float)(n % WW) + 0.5f) * s;
      float ancy = ((float)(n / WW) + 0.5f) * s;
      const float* tp = gt_pts + (size_t)(b * NGRP + g) * 6;
      float gx[3], gy[3], qx[3], qy[3];
#pragma unroll
      for (int j = 0; j < 3; ++j) {
        gx[j] = (tp[j * 2 + 0] - ancx) / s;
        gy[j] = (tp[j * 2 + 1] - ancy) / s;
        qx[j] = pred_reg[((size_t)(b * 6) + (j * 2 + 0)) * NPIX + n];
        qy[j] = pred_reg[((size_t)(b * 6) + (j * 2 + 1)) * NPIX + n];
      }
      float dx0 = qx[0] - gx[0], dy0 = qy[0] - gy[0];
      acc = dx0 * dx0 + dy0 * dy0;                       // LP0 = 1
      float dmat[2][2];
#pragma unroll
      for (int r = 0; r < 2; ++r)
#pragma unroll
        for (int c = 0; c < 2; ++c) {
          float dx = qx[r + 1] - gx[c + 1], dy = qy[r + 1] - gy[c + 1];
          dmat[r][c] = sqrtf(dx * dx + dy * dy + 1e-12f);
        }
      acc += fminf(dmat[0][0], dmat[0][1]) + fminf(dmat[1][0], dmat[1][1])
           + fminf(dmat[0][0], dmat[1][0]) + fminf(dmat[0][1], dmat[1][1]);  // LCD = 1
    }
  }
  r0[threadIdx.x] = acc;
  __syncthreads();
  for (int off = 128; off > 0; off >>= 1) {
    if (threadIdx.x < off) r0[threadIdx.x] += r0[threadIdx.x + off];
    __syncthreads();
  }
  if (threadIdx.x == 0) atomicAdd(&out[0], r0[0]);
}

extern "C" void kernel_launch(void* const* d_in, const int* in_sizes, int n_in,
                              void* d_out, int out_size, void* d_ws, size_t ws_size,
                              hipStream_t stream) {
  const float* pred_reg = (const float*)d_in[0];
  const float* pred_obj = (const float*)d_in[1];
  const float* pred_cls = (const float*)d_in[2];
  const float* gt_pts   = (const float*)d_in[3];
  const int*   gt_lbl   = (const int*)d_in[4];
  const int*   stride_p = (const int*)d_in[5];
  float* out = (float*)d_out;

  unsigned long long* sel_key = (unsigned long long*)d_ws;            // 24576 * 8B
  unsigned long long* cellkey = sel_key + (size_t)NB * NGRP * KCAP;   // 147456 * 8B
  (void)in_sizes; (void)n_in; (void)out_size; (void)ws_size;

  k_init<<<(NB * NPIX + 255) / 256, 256, 0, stream>>>(cellkey, out);
  k_select<<<NB * NGRP, 256, 0, stream>>>(gt_pts, stride_p, sel_key, cellkey);
  k_objcls<<<(NB * NPIX + 255) / 256, 256, 0, stream>>>(pred_obj, pred_cls, gt_lbl, cellkey, out);
  k_reg<<<(NB * NGRP * KCAP + 255) / 256, 256, 0, stream>>>(pred_reg, gt_pts, stride_p, sel_key, out);
}